// GNN_85134841741882
// MI455X (gfx1250) — compile-verified
//
#include <hip/hip_runtime.h>
#include <math.h>

#define N_NODES 50000
#define N_EDGES 800000
#define CH      256          // D_IN == D_HID == D_OUT

typedef __attribute__((ext_vector_type(2))) float v2f;
typedef __attribute__((ext_vector_type(8))) float v8f;

// ---------------------------------------------------------------------------
// 1) Degree: deg[dst] += 1 per edge (atomic, L2-resident, no-return)
// ---------------------------------------------------------------------------
__global__ __launch_bounds__(256) void deg_kernel(const int* __restrict__ dst,
                                                  float* __restrict__ cnt, int nE) {
    int e = blockIdx.x * 256 + threadIdx.x;
    if (e < nE) atomicAdd(&cnt[dst[e]], 1.0f);
}

// inv[i] = 1 / max(deg, 1)
__global__ __launch_bounds__(256) void inv_kernel(const float* __restrict__ cnt,
                                                  float* __restrict__ inv, int n) {
    int i = blockIdx.x * 256 + threadIdx.x;
    if (i < n) inv[i] = 1.0f / fmaxf(cnt[i], 1.0f);
}

// ---------------------------------------------------------------------------
// 2) Edge scatter: agg[dst] += feat[src]  (256 threads == 256 channels; the
//    gather of feat[src] is a fully coalesced 1KB row read, the atomic adds
//    land in a 51.2MB accumulator that stays resident in the 192MB L2)
// ---------------------------------------------------------------------------
#define EDGES_PER_BLOCK 16
__global__ __launch_bounds__(256) void scatter_kernel(const float* __restrict__ feat,
                                                      const int* __restrict__ src,
                                                      const int* __restrict__ dst,
                                                      float* __restrict__ agg, int nE) {
    int t  = threadIdx.x;                       // channel
    int e0 = blockIdx.x * EDGES_PER_BLOCK;
    #pragma unroll
    for (int i = 0; i < EDGES_PER_BLOCK; ++i) {
        int e = e0 + i;
        if (e < nE) {
            int s = src[e], d = dst[e];
            float v = feat[(size_t)s * CH + t];
            atomicAdd(&agg[(size_t)d * CH + t], v);
        }
    }
}

// ---------------------------------------------------------------------------
// 3) Fused SAGE GEMM via V_WMMA_F32_16X16X4_F32:
//    out = maybe_relu( (agg*inv) @ Wl^T + bl + xin @ Wr^T )
//    Block = 256 threads = 8 waves; macro-tile 128 rows x 16 cols.
//    Wl/Wr rows for this col-tile (16x256 each, 32KB) staged in LDS.
//    Per wave: one 16x16 f32 accumulator, 64 K-steps, 2 WMMAs/step.
//
//    A layout (16x4 f32, ISA 7.12.2): lane = {half=lane>>4, m=lane&15}
//      vgpr0/1 = A[m][kb + 2*half + {0,1}]   -> one float2 load
//    B layout (4x16 f32): vgpr0/1 = B[2*half + {0,1}][m] = W[col0+m][kb+2*half+{0,1}]
//    C/D layout: c[r] = out[row0 + r + 8*half][col0 + m]
// ---------------------------------------------------------------------------
__global__ __launch_bounds__(256) void sage_gemm_kernel(
        const float* __restrict__ agg, const float* __restrict__ inv,
        const float* __restrict__ xin,
        const float* __restrict__ Wl, const float* __restrict__ bl,
        const float* __restrict__ Wr,
        float* __restrict__ out, int nRows, int doRelu) {
    __shared__ float sWl[16 * CH];
    __shared__ float sWr[16 * CH];

    const int col0 = blockIdx.y * 16;

    // stage W tiles into LDS (1024 float4 each)
    {
        const float4* gl = reinterpret_cast<const float4*>(Wl + (size_t)col0 * CH);
        const float4* gr = reinterpret_cast<const float4*>(Wr + (size_t)col0 * CH);
        float4* sl = reinterpret_cast<float4*>(sWl);
        float4* sr = reinterpret_cast<float4*>(sWr);
        for (int i = threadIdx.x; i < (16 * CH) / 4; i += 256) {
            sl[i] = gl[i];
            sr[i] = gr[i];
        }
    }
    __syncthreads();

    const int wave = threadIdx.x >> 5;
    const int lane = threadIdx.x & 31;
    const int half = lane >> 4;        // 0: K+0/1, 1: K+2/3
    const int m    = lane & 15;        // A row within tile / B col within tile

    const int row0 = blockIdx.x * 128 + wave * 16;
    int rA = row0 + m;
    if (rA > nRows - 1) rA = nRows - 1;          // clamp for safe loads; stores guarded

    const float  sc     = inv[rA];
    const float* aggRow = agg + (size_t)rA * CH;
    const float* xRow   = xin + (size_t)rA * CH;
    const float* wlRow  = sWl + m * CH;
    const float* wrRow  = sWr + m * CH;

    v8f c = {};
    #pragma unroll 4
    for (int kb = 0; kb < CH; kb += 4) {
        const int off = kb + 2 * half;
        float2 a1 = *reinterpret_cast<const float2*>(aggRow + off);
        float2 a2 = *reinterpret_cast<const float2*>(xRow + off);
        float2 b1 = *reinterpret_cast<const float2*>(wlRow + off);
        float2 b2 = *reinterpret_cast<const float2*>(wrRow + off);

        v2f A1 = { a1.x * sc, a1.y * sc };   // fold mean division into A
        v2f A2 = { a2.x, a2.y };
        v2f B1 = { b1.x, b1.y };
        v2f B2 = { b2.x, b2.y };

        c = __builtin_amdgcn_wmma_f32_16x16x4_f32(false, A1, false, B1,
                                                  (short)0, c, false, false);
        c = __builtin_amdgcn_wmma_f32_16x16x4_f32(false, A2, false, B2,
                                                  (short)0, c, false, false);
    }

    const float bias = bl[col0 + m];
    float* outp = out + (size_t)(row0 + 8 * half) * CH + col0 + m;

    if (row0 + 16 <= nRows) {
        // wave-uniform fast path: single scalar branch, no exec-mask churn
        #pragma unroll
        for (int r = 0; r < 8; ++r) {
            float v = c[r] + bias;
            if (doRelu) v = fmaxf(v, 0.0f);
            outp[(size_t)r * CH] = v;
        }
    } else {
        #pragma unroll
        for (int r = 0; r < 8; ++r) {
            const int row = row0 + r + 8 * half;
            if (row < nRows) {
                float v = c[r] + bias;
                if (doRelu) v = fmaxf(v, 0.0f);
                outp[(size_t)r * CH] = v;
            }
        }
    }
}

// ---------------------------------------------------------------------------
// 4) logits[i] = dot(h[mention], h[i]); one wave per node. Each lane owns a
//    contiguous 8-float chunk (two b128 loads), then wave32 shfl reduce.
// ---------------------------------------------------------------------------
__global__ __launch_bounds__(256) void logits_kernel(const float* __restrict__ h,
                                                     const int* __restrict__ mention,
                                                     float* __restrict__ logits, int n) {
    __shared__ float sm[CH];
    const int mrow = mention[0];
    for (int t = threadIdx.x; t < CH; t += 256) sm[t] = h[(size_t)mrow * CH + t];
    __syncthreads();

    const int wave = threadIdx.x >> 5;
    const int lane = threadIdx.x & 31;
    const int i = blockIdx.x * 8 + wave;
    if (i >= n) return;

    const float4* row4 = reinterpret_cast<const float4*>(h + (size_t)i * CH) + lane * 2;
    const float4* sm4  = reinterpret_cast<const float4*>(sm) + lane * 2;
    float4 r0 = row4[0], r1 = row4[1];
    float4 m0 = sm4[0],  m1 = sm4[1];
    float s = r0.x * m0.x + r0.y * m0.y + r0.z * m0.z + r0.w * m0.w
            + r1.x * m1.x + r1.y * m1.y + r1.z * m1.z + r1.w * m1.w;
    #pragma unroll
    for (int o = 16; o > 0; o >>= 1) s += __shfl_xor(s, o, 32);
    if (lane == 0) logits[i] = s;
}

// ---------------------------------------------------------------------------
// 5) softmax over N logits: partial max -> max -> partial sumexp -> sum -> z
// ---------------------------------------------------------------------------
#define RED_BLOCKS 128
__global__ __launch_bounds__(256) void max_partial_kernel(const float* __restrict__ x,
                                                          float* __restrict__ red, int n) {
    __shared__ float sm[256];
    float v = -INFINITY;
    for (int i = blockIdx.x * 256 + threadIdx.x; i < n; i += gridDim.x * 256)
        v = fmaxf(v, x[i]);
    sm[threadIdx.x] = v; __syncthreads();
    for (int s = 128; s > 0; s >>= 1) {
        if (threadIdx.x < s) sm[threadIdx.x] = fmaxf(sm[threadIdx.x], sm[threadIdx.x + s]);
        __syncthreads();
    }
    if (threadIdx.x == 0) red[blockIdx.x] = sm[0];
}

__global__ __launch_bounds__(256) void max_final_kernel(float* __restrict__ red) {
    __shared__ float sm[256];
    float v = (threadIdx.x < RED_BLOCKS) ? red[threadIdx.x] : -INFINITY;
    sm[threadIdx.x] = v; __syncthreads();
    for (int s = 128; s > 0; s >>= 1) {
        if (threadIdx.x < s) sm[threadIdx.x] = fmaxf(sm[threadIdx.x], sm[threadIdx.x + s]);
        __syncthreads();
    }
    if (threadIdx.x == 0) red[255] = sm[0];       // gmax slot
}

__global__ __launch_bounds__(256) void sum_partial_kernel(const float* __restrict__ x,
                                                          float* __restrict__ red, int n) {
    __shared__ float sm[256];
    const float gmax = red[255];
    float v = 0.0f;
    for (int i = blockIdx.x * 256 + threadIdx.x; i < n; i += gridDim.x * 256)
        v += expf(x[i] - gmax);
    sm[threadIdx.x] = v; __syncthreads();
    for (int s = 128; s > 0; s >>= 1) {
        if (threadIdx.x < s) sm[threadIdx.x] += sm[threadIdx.x + s];
        __syncthreads();
    }
    if (threadIdx.x == 0) red[256 + blockIdx.x] = sm[0];
}

__global__ __launch_bounds__(256) void sum_final_kernel(float* __restrict__ red) {
    __shared__ float sm[256];
    float v = (threadIdx.x < RED_BLOCKS) ? red[256 + threadIdx.x] : 0.0f;
    sm[threadIdx.x] = v; __syncthreads();
    for (int s = 128; s > 0; s >>= 1) {
        if (threadIdx.x < s) sm[threadIdx.x] += sm[threadIdx.x + s];
        __syncthreads();
    }
    if (threadIdx.x == 0) red[511] = sm[0];       // gsum slot
}

__global__ __launch_bounds__(256) void softmax_write_kernel(const float* __restrict__ logits,
                                                            const float* __restrict__ red,
                                                            float* __restrict__ z, int n) {
    int i = blockIdx.x * 256 + threadIdx.x;
    if (i < n) z[i] = expf(logits[i] - red[255]) / red[511];
}

// ---------------------------------------------------------------------------
// launch
// ---------------------------------------------------------------------------
extern "C" void kernel_launch(void* const* d_in, const int* in_sizes, int n_in,
                              void* d_out, int out_size, void* d_ws, size_t ws_size,
                              hipStream_t stream) {
    const float* x       = (const float*)d_in[0];
    const int*   eidx    = (const int*)d_in[1];    // [2][E]: src then dst
    const int*   mention = (const int*)d_in[2];
    const float* W1_l    = (const float*)d_in[3];
    const float* b1_l    = (const float*)d_in[4];
    const float* W1_r    = (const float*)d_in[5];
    const float* W2_l    = (const float*)d_in[6];
    const float* b2_l    = (const float*)d_in[7];
    const float* W2_r    = (const float*)d_in[8];

    const int* src = eidx;
    const int* dst = eidx + N_EDGES;

    // workspace layout (floats)
    float* ws     = (float*)d_ws;
    float* agg    = ws;                                   // N*CH (reused by both layers)
    float* h1     = agg + (size_t)N_NODES * CH;           // N*CH
    float* h2     = h1  + (size_t)N_NODES * CH;           // N*CH
    float* cnt    = h2  + (size_t)N_NODES * CH;           // N
    float* inv    = cnt + N_NODES;                        // N
    float* logit  = inv + N_NODES;                        // N
    float* red    = logit + N_NODES;                      // 512

    const int NB_N = (N_NODES + 255) / 256;
    const int NB_E = (N_EDGES + 255) / 256;
    const int NB_SCAT = (N_EDGES + EDGES_PER_BLOCK - 1) / EDGES_PER_BLOCK;
    const dim3 gemm_grid((N_NODES + 127) / 128, CH / 16);

    // degree + inverse (graph is identical for both layers)
    hipMemsetAsync(cnt, 0, N_NODES * sizeof(float), stream);
    deg_kernel<<<NB_E, 256, 0, stream>>>(dst, cnt, N_EDGES);
    inv_kernel<<<NB_N, 256, 0, stream>>>(cnt, inv, N_NODES);

    // ---- layer 1 ----
    hipMemsetAsync(agg, 0, (size_t)N_NODES * CH * sizeof(float), stream);
    scatter_kernel<<<NB_SCAT, 256, 0, stream>>>(x, src, dst, agg, N_EDGES);
    sage_gemm_kernel<<<gemm_grid, 256, 0, stream>>>(agg, inv, x, W1_l, b1_l, W1_r,
                                                    h1, N_NODES, /*relu=*/1);

    // ---- layer 2 ----
    hipMemsetAsync(agg, 0, (size_t)N_NODES * CH * sizeof(float), stream);
    scatter_kernel<<<NB_SCAT, 256, 0, stream>>>(h1, src, dst, agg, N_EDGES);
    sage_gemm_kernel<<<gemm_grid, 256, 0, stream>>>(agg, inv, h1, W2_l, b2_l, W2_r,
                                                    h2, N_NODES, /*relu=*/0);

    // ---- logits + softmax ----
    logits_kernel<<<(N_NODES + 7) / 8, 256, 0, stream>>>(h2, mention, logit, N_NODES);
    max_partial_kernel<<<RED_BLOCKS, 256, 0, stream>>>(logit, red, N_NODES);
    max_final_kernel<<<1, 256, 0, stream>>>(red);
    sum_partial_kernel<<<RED_BLOCKS, 256, 0, stream>>>(logit, red, N_NODES);
    sum_final_kernel<<<1, 256, 0, stream>>>(red);
    softmax_write_kernel<<<NB_N, 256, 0, stream>>>(logit, red, (float*)d_out, N_NODES);
}